// RelposEncoding_50903952392794
// MI455X (gfx1250) — compile-verified
//
#include <hip/hip_runtime.h>
#include <stdint.h>

#define BB 4
#define SS 512
#define DHEAD 64
#define POSITIONS 441
#define NFEAT 8

typedef float v4f __attribute__((ext_vector_type(4)));
typedef float v2f __attribute__((ext_vector_type(2)));

__global__ __launch_bounds__(256) void relpos_gather_kernel(
    const float* __restrict__ feats,
    const float* __restrict__ keys_emb,
    const float* __restrict__ values_emb,
    const int*   __restrict__ index_map,
    const int*   __restrict__ entity_type,
    float* __restrict__ keys_out,
    float* __restrict__ values_out)
{
    __shared__ v2f s_pos[SS];   // (x, y) per key slot of this batch
    __shared__ int s_et[SS];    // entity type per key slot

    const int bi  = blockIdx.x;        // 0 .. B*S-1  (one block per (b, i))
    const int b   = bi >> 9;           // / 512
    const int i   = bi & (SS - 1);
    const int tid = threadIdx.x;

    // ---- Stage this batch's positions + entity types into LDS via the
    // ---- gfx1250 async global->LDS path (tracked by ASYNCcnt).
    #pragma unroll
    for (int t = tid; t < SS; t += 256) {
        const int g = index_map[b * SS + t];

        // async copy 8 bytes (x,y) straight into LDS
        {
            unsigned lds_off = (unsigned)(uintptr_t)(&s_pos[t]);
            unsigned long long gaddr =
                (unsigned long long)(uintptr_t)(feats + (size_t)g * NFEAT);
            asm volatile("global_load_async_to_lds_b64 %0, %1, off"
                         :: "v"(lds_off), "v"(gaddr) : "memory");
        }
        // async copy 4 bytes (entity type) into LDS
        {
            unsigned lds_off = (unsigned)(uintptr_t)(&s_et[t]);
            unsigned long long gaddr =
                (unsigned long long)(uintptr_t)(entity_type + g);
            asm volatile("global_load_async_to_lds_b32 %0, %1, off"
                         :: "v"(lds_off), "v"(gaddr) : "memory");
        }
    }
    asm volatile("s_wait_asynccnt 0" ::: "memory");
    __syncthreads();

    const v2f  pi  = s_pos[i];

    const int lane16 = tid & 15;   // which float4 slice of the 64-float row
    const int jg     = tid >> 4;   // 16 j-groups per block
    const size_t rowbase = (size_t)bi * (SS * DHEAD);

    #pragma unroll 2
    for (int j = jg; j < SS; j += 16) {
        const v2f pj = s_pos[j];
        // bucketize relative position (round-half-even, clamp to [-10, 10])
        float dx = fminf(fmaxf(rintf(pj.x - pi.x), -10.0f), 10.0f);
        float dy = fminf(fmaxf(rintf(pj.y - pi.y), -10.0f), 10.0f);
        int idx  = (int)(dx + 10.0f) + 21 * (int)(dy + 10.0f);       // 0..440
        int pidx = idx + s_et[j] * POSITIONS;                        // 0..1763

        // coalesced 256B row gathers (L2-resident tables)
        v4f kv = *((const v4f*)(keys_emb   + (size_t)idx  * DHEAD) + lane16);
        v4f vv = *((const v4f*)(values_emb + (size_t)pidx * DHEAD) + lane16);

        // streaming non-temporal stores: don't pollute L2 with the 512MB output
        v4f* kdst = (v4f*)(keys_out   + rowbase + (size_t)j * DHEAD) + lane16;
        v4f* vdst = (v4f*)(values_out + rowbase + (size_t)j * DHEAD) + lane16;
        __builtin_nontemporal_store(kv, kdst);
        __builtin_nontemporal_store(vv, vdst);
    }
}

extern "C" void kernel_launch(void* const* d_in, const int* in_sizes, int n_in,
                              void* d_out, int out_size, void* d_ws, size_t ws_size,
                              hipStream_t stream) {
    const float* feats       = (const float*)d_in[0];
    const float* keys_emb    = (const float*)d_in[1];
    const float* values_emb  = (const float*)d_in[2];
    const int*   index_map   = (const int*)d_in[3];
    const int*   entity_type = (const int*)d_in[4];

    float* keys_out   = (float*)d_out;
    float* values_out = keys_out + (size_t)BB * SS * SS * DHEAD;

    dim3 grid(BB * SS);   // 2048 blocks: one per (b, i)
    dim3 block(256);      // 8 waves; 16 lanes per key row
    relpos_gather_kernel<<<grid, block, 0, stream>>>(
        feats, keys_emb, values_emb, index_map, entity_type,
        keys_out, values_out);
}